// TopKGate_13984413516278
// MI455X (gfx1250) — compile-verified
//
#include <hip/hip_runtime.h>
#include <math.h>

typedef __attribute__((ext_vector_type(2))) float v2f;
typedef __attribute__((ext_vector_type(8))) float v8f;
typedef __attribute__((ext_vector_type(4))) int   v4i;

#define TOKENS      8192
#define MODEL_DIM   4096
#define NUM_EXPERTS 64

#define KCHUNK      128                    // K per staged chunk
#define NCHUNK      (MODEL_DIM / KCHUNK)   // 32
#define ASTRIDE     (KCHUNK + 4)           // 132 floats: stride % 64 == 4 -> conflict-free b64 reads

// ---- async global->LDS helpers ----
__device__ __forceinline__ void async_copy_b128(const float* gsrc, float* ldst) {
#if __has_builtin(__builtin_amdgcn_global_load_async_to_lds_b128)
    __builtin_amdgcn_global_load_async_to_lds_b128(
        (__attribute__((address_space(1))) v4i*)(void*)(const void*)gsrc,
        (__attribute__((address_space(3))) v4i*)(void*)ldst,
        /*offset=*/0, /*cpol=*/0);
#else
    *(float4*)ldst = *(const float4*)gsrc;
#endif
}

template <int N>
__device__ __forceinline__ void wait_async() {
#if __has_builtin(__builtin_amdgcn_s_wait_asynccnt)
    __builtin_amdgcn_s_wait_asynccnt(N);
#else
    asm volatile("s_wait_asynccnt %0" :: "i"(N) : "memory");
#endif
}

// Block: 128 threads = 4 wave32s. Block computes 16 tokens x 64 experts.
// Wave w owns the 16x16 tile for experts [16w, 16w+16).
// x tile is staged cooperatively into LDS (async, double-buffered); A fragments
// are read from LDS, B fragments stream from global (W is L2-resident).
__global__ __launch_bounds__(128) void moe_top2_gate_kernel(
    const float* __restrict__ x,   // [TOKENS, MODEL_DIM]
    const float* __restrict__ W,   // [NUM_EXPERTS, MODEL_DIM]
    float* __restrict__ out)       // [4 * TOKENS]: idx0, idx1, g0, g1
{
    __shared__ float ldsA[2 * 16 * ASTRIDE];      // double-buffered A tile (~16.5 KB)
    __shared__ float logits[16 * NUM_EXPERTS];    // [token][expert], 4 KB

    const int tid  = threadIdx.x;
    const int wave = tid >> 5;
    const int lane = tid & 31;
    const int hi   = lane >> 4;    // 0: lanes 0-15, 1: lanes 16-31
    const int l16  = lane & 15;
    const int tok0 = blockIdx.x * 16;

    // Cooperative stage of one K-chunk: 16 rows x 128 floats = 8 KB.
    // Each wave instruction copies one full row (32 lanes x 16B = 512B contiguous).
    // 4 instructions per wave per chunk -> ASYNCcnt watermark of 4.
    auto stage_chunk = [&](int kbase, int buf) {
#pragma unroll
        for (int j = 0; j < 4; ++j) {
            int idx = tid + 128 * j;        // 0..511 float4 slots
            int row = idx >> 5;             // 0..15
            int c4  = (idx & 31) << 2;      // 0,4,...,124
            async_copy_b128(x + (size_t)(tok0 + row) * MODEL_DIM + kbase + c4,
                            ldsA + buf * (16 * ASTRIDE) + row * ASTRIDE + c4);
        }
    };

    // B-fragment source: W row (expert 16*wave + l16), K = k + 2*hi + {0,1}
    const float* brow = W + (size_t)(wave * 16 + l16) * MODEL_DIM + 2 * hi;
    // A-fragment source in LDS: row l16, same K striping (ISA 16x4 f32 layout)
    const float* alds = ldsA + l16 * ASTRIDE + 2 * hi;

    stage_chunk(0, 0);                      // prologue: chunk 0 -> buf 0

    v8f acc = {};
    for (int c = 0; c < NCHUNK; ++c) {
        const int buf = c & 1;
        if (c + 1 < NCHUNK) {
            stage_chunk((c + 1) * KCHUNK, buf ^ 1);
            wait_async<4>();                // chunk c's 4 wave-instructions done
        } else {
            wait_async<0>();
        }
        __syncthreads();                    // chunk c visible to all waves

        const float* a = alds + buf * (16 * ASTRIDE);
        const float* b = brow + c * KCHUNK;
#pragma unroll 4
        for (int kk = 0; kk < KCHUNK; kk += 4) {
            v2f av = *(const v2f*)(a + kk);         // ds_load_b64, bank-conflict-free
            v2f bv = *(const v2f*)(b + kk);         // global_load_b64, coalesced
            acc = __builtin_amdgcn_wmma_f32_16x16x4_f32(
                false, av, false, bv, (short)0, acc, false, false);
        }
        __syncthreads();                    // safe to overwrite buf next iteration
    }

    // C/D layout: VGPR r, lanes 0-15 -> (M=r, N=l16); lanes 16-31 -> (M=r+8, N=l16)
    const int e = wave * 16 + l16;
#pragma unroll
    for (int r = 0; r < 8; ++r) {
        logits[(r + 8 * hi) * NUM_EXPERTS + e] = acc[r];
    }
    __syncthreads();

    // One thread per token: branchless top-2, stable softmax gather, renormalize.
    if (tid < 16) {
        const float* lrow = logits + tid * NUM_EXPERTS;
        float m1 = -INFINITY, m2 = -INFINITY;
        int   i1 = 0,         i2 = 0;
        for (int ei = 0; ei < NUM_EXPERTS; ++ei) {
            float v   = lrow[ei];
            bool  g1b = v > m1;
            bool  g2b = v > m2;
            float nm2 = g1b ? m1 : (g2b ? v  : m2);
            int   ni2 = g1b ? i1 : (g2b ? ei : i2);
            m1 = g1b ? v  : m1;
            i1 = g1b ? ei : i1;
            m2 = nm2;
            i2 = ni2;
        }
        float s = 0.f;
        for (int ei = 0; ei < NUM_EXPERTS; ++ei) s += __expf(lrow[ei] - m1);
        float g1 = 1.0f / s;                 // exp(m1 - m1) / s
        float g2 = __expf(m2 - m1) / s;
        float denom = fmaxf(g1 + g2, 1.1920929e-07f);  // FLT_EPSILON clamp
        const int t = tok0 + tid;
        out[t]              = (float)i1;
        out[TOKENS + t]     = (float)i2;
        out[2 * TOKENS + t] = g1 / denom;
        out[3 * TOKENS + t] = g2 / denom;
    }
}

extern "C" void kernel_launch(void* const* d_in, const int* in_sizes, int n_in,
                              void* d_out, int out_size, void* d_ws, size_t ws_size,
                              hipStream_t stream) {
    (void)in_sizes; (void)n_in; (void)out_size; (void)d_ws; (void)ws_size;
    const float* x = (const float*)d_in[0];
    const float* W = (const float*)d_in[1];
    float* out     = (float*)d_out;

    dim3 grid(TOKENS / 16);   // 512 blocks
    dim3 block(128);          // 4 wave32s
    hipLaunchKernelGGL(moe_top2_gate_kernel, grid, block, 0, stream, x, W, out);
}